// GetAffs_71708773974139
// MI455X (gfx1250) — compile-verified
//
#include <hip/hip_runtime.h>
#include <stdint.h>

// Affinity-from-segmentation stencil for MI455X (gfx1250).
// Memory-bound: 32MB in + 256MB out -> ~12.4us floor at 23.3 TB/s.
// CDNA5 paths used:
//   - global_load_async_to_lds_b128 (ASYNCcnt) to stage tile + 27-deep
//     top/left halo into LDS with no VGPR staging, 512B/wave-request.
//   - s_wait_asynccnt 0 before the workgroup barrier.
//   - non-temporal b128 stores so the 256MB output stream does not evict
//     the 32MB input working set (halo rows re-read ~3.3x) from L2.

namespace {
constexpr int TX       = 128;            // output tile width
constexpr int TY       = 16;             // output tile height
constexpr int HALO_T   = 27;             // top halo (max |offset|)
constexpr int HALO_L   = 28;             // left halo, padded 27->28: 16B alignment
constexpr int LDS_W    = TX + HALO_L;    // 156 floats (624B, multiple of 16)
constexpr int LDS_H    = TY + HALO_T;    // 43
constexpr int LDS_N    = LDS_W * LDS_H;  // 6708 floats = 26.8 KB
constexpr int CHUNKS   = LDS_W / 4;      // 39 float4 chunks per LDS row
constexpr int IMG_H    = 1024;
constexpr int IMG_W    = 1024;
constexpr int NTHREADS = 256;            // 8 wave32s
}

typedef float v4f __attribute__((ext_vector_type(4)));

__global__ __launch_bounds__(NTHREADS)
void affs_stencil_kernel(const float* __restrict__ seg, float* __restrict__ out)
{
    __shared__ float tile[LDS_N];

    const int tid = threadIdx.x;
    const int tx  = tid & 31;   // lane
    const int tyq = tid >> 5;   // wave/row-group 0..7
    const int j0  = blockIdx.x * TX;
    const int i0  = blockIdx.y * TY;
    const int b   = blockIdx.z;

    const float* segB = seg + (size_t)b * (IMG_H * IMG_W);

    // ---------------- Phase 1: async global -> LDS tile fill -----------------
    // 43 rows x 39 aligned float4 chunks. j0 is a multiple of 128 and HALO_L=28,
    // so every chunk is 16B-aligned in global AND LDS, and chunks are OOB only
    // as a whole (clamped chunks land in LDS cells that are consumed solely
    // under a false in-bounds predicate).
    const uint32_t lds_base = (uint32_t)(uintptr_t)&tile[0];
    for (int r = tyq; r < LDS_H; r += 8) {
        int gi = i0 + r - HALO_T; gi = gi < 0 ? 0 : gi;          // gi < IMG_H always
        const float* grow = segB + (size_t)gi * IMG_W;           // row base (gj = 0)
        const uint32_t lrow = lds_base + (uint32_t)(r * LDS_W) * 4u;
        for (int ch = tx; ch < CHUNKS; ch += 32) {
            const int gj = j0 - HALO_L + 4 * ch;                 // multiple of 4
            const float* gp = (gj < 0) ? grow : (grow + gj);     // chunk-level clamp
            const uint64_t gaddr = (uint64_t)(uintptr_t)gp;
            const uint32_t laddr = lrow + (uint32_t)ch * 16u;
            asm volatile("global_load_async_to_lds_b128 %0, %1, off"
                         :: "v"(laddr), "v"(gaddr)
                         : "memory");
        }
    }
    asm volatile("s_wait_asynccnt 0" ::: "memory");   // this wave's fills landed
    __syncthreads();                                   // all waves' fills landed

    // ---------------- Phase 2: 8 shift-compares per pixel --------------------
    const size_t plane    = (size_t)IMG_H * IMG_W;
    const size_t offPitch = (size_t)gridDim.z * plane;  // stride between offset planes
    float* outB = out + (size_t)b * plane;

    const int dvals[4] = {1, 3, 9, 27};

    #pragma unroll
    for (int rr = 0; rr < 2; ++rr) {
        const int rowInTile = tyq + rr * 8;
        const int r  = HALO_T + rowInTile;     // LDS row of center
        const int gi = i0 + rowInTile;
        const int gj = j0 + 4 * tx;
        const int c  = HALO_L + 4 * tx;        // LDS col of center (16B aligned)

        const float* rowp = &tile[(size_t)r * LDS_W];
        const float c0 = rowp[c + 0];
        const float c1 = rowp[c + 1];
        const float c2 = rowp[c + 2];
        const float c3 = rowp[c + 3];

        float* orow = outB + (size_t)gi * IMG_W + gj;

        #pragma unroll
        for (int k = 0; k < 4; ++k) {
            const int d = dvals[k];

            // Offset (-d, 0): vertical shift -> output plane 2k.
            {
                const float* sp = &tile[(size_t)(r - d) * LDS_W];
                const bool ib = (gi - d) >= 0;       // zero-padding above image
                const float s0 = ib ? sp[c + 0] : 0.0f;
                const float s1 = ib ? sp[c + 1] : 0.0f;
                const float s2 = ib ? sp[c + 2] : 0.0f;
                const float s3 = ib ? sp[c + 3] : 0.0f;
                v4f v;
                v.x = (s0 == c0) ? 1.0f : 0.0f;
                v.y = (s1 == c1) ? 1.0f : 0.0f;
                v.z = (s2 == c2) ? 1.0f : 0.0f;
                v.w = (s3 == c3) ? 1.0f : 0.0f;
                __builtin_nontemporal_store(
                    v, reinterpret_cast<v4f*>(orow + (size_t)(2 * k) * offPitch));
            }

            // Offset (0, -d): horizontal shift -> output plane 2k+1.
            {
                const float s0 = (gj + 0 - d) >= 0 ? rowp[c + 0 - d] : 0.0f;
                const float s1 = (gj + 1 - d) >= 0 ? rowp[c + 1 - d] : 0.0f;
                const float s2 = (gj + 2 - d) >= 0 ? rowp[c + 2 - d] : 0.0f;
                const float s3 = (gj + 3 - d) >= 0 ? rowp[c + 3 - d] : 0.0f;
                v4f v;
                v.x = (s0 == c0) ? 1.0f : 0.0f;
                v.y = (s1 == c1) ? 1.0f : 0.0f;
                v.z = (s2 == c2) ? 1.0f : 0.0f;
                v.w = (s3 == c3) ? 1.0f : 0.0f;
                __builtin_nontemporal_store(
                    v, reinterpret_cast<v4f*>(orow + (size_t)(2 * k + 1) * offPitch));
            }
        }
    }
}

extern "C" void kernel_launch(void* const* d_in, const int* in_sizes, int n_in,
                              void* d_out, int out_size, void* d_ws, size_t ws_size,
                              hipStream_t stream) {
    const float* seg = (const float*)d_in[0];
    float* out = (float*)d_out;

    const int B = in_sizes[0] / (IMG_H * IMG_W);   // 8 for the reference shapes

    dim3 grid(IMG_W / TX, IMG_H / TY, B);          // 8 x 64 x 8 = 4096 blocks
    affs_stencil_kernel<<<grid, NTHREADS, 0, stream>>>(seg, out);
}